// GCNEncoder_1683627180498
// MI455X (gfx1250) — compile-verified
//
#include <hip/hip_runtime.h>
#include <hip/hip_bf16.h>

typedef float v2f __attribute__((ext_vector_type(2)));
typedef float v8f __attribute__((ext_vector_type(8)));

// ---------------------------------------------------------------------------
// Degree / normalization
// ---------------------------------------------------------------------------
__global__ void deg_init_kernel(float* __restrict__ deg, int n) {
    int i = blockIdx.x * blockDim.x + threadIdx.x;
    if (i < n) deg[i] = 1.0f;  // self-loop weight 1.0
}

__global__ void deg_scatter_kernel(const long long* __restrict__ ei,
                                   const float* __restrict__ w,
                                   float* __restrict__ deg, int E) {
    int e = blockIdx.x * blockDim.x + threadIdx.x;
    if (e < E) {
        int c = (int)ei[(long long)E + e];  // col
        atomicAdd(&deg[c], w[e]);
    }
}

__global__ void dinv_kernel(const float* __restrict__ deg,
                            float* __restrict__ dinv, int n) {
    int i = blockIdx.x * blockDim.x + threadIdx.x;
    if (i < n) {
        float d = deg[i];
        dinv[i] = (d > 0.0f) ? rsqrtf(fmaxf(d, 1e-12f)) : 0.0f;
    }
}

// ---------------------------------------------------------------------------
// GEMM1: H[M,64] = X[M,11] @ W1[11,64]   (K padded 11 -> 12, 3 WMMA steps)
// 256 threads = 8 waves; wave w -> m-tile (w>>2), n-tile (w&3); block = 32 rows
// ---------------------------------------------------------------------------
__global__ void gemm1_wmma_kernel(const float* __restrict__ X,
                                  const float* __restrict__ W1,
                                  float* __restrict__ H, int M) {
    __shared__ float sA[32 * 12];
    __shared__ float sB[12 * 64];
    const int m0 = blockIdx.x * 32;
    const int t = threadIdx.x;

    for (int i = t; i < 32 * 12; i += 256) {
        int r = i / 12, c = i - r * 12;
        int row = m0 + r; if (row >= M) row = M - 1;   // clamp (exact fit anyway)
        sA[i] = (c < 11) ? X[row * 11 + c] : 0.0f;
    }
    for (int i = t; i < 12 * 64; i += 256) {
        int k = i >> 6;
        sB[i] = (k < 11) ? W1[i] : 0.0f;               // W1 row-major [11,64]
    }
    __syncthreads();

    const int wave = t >> 5, lane = t & 31;
    const int mt = wave >> 2, nt = wave & 3;
    const int half = lane >> 4, l = lane & 15;
    const int ar = mt * 16 + l;                        // A row within tile pair
    const int n0 = nt * 16;

    v8f acc = {};
    #pragma unroll
    for (int kk = 0; kk < 12; kk += 4) {
        int ka = kk + half * 2;                        // K for VGPR0 per ISA layout
        v2f a, b;
        a.x = sA[ar * 12 + ka];
        a.y = sA[ar * 12 + ka + 1];
        b.x = sB[ka * 64 + n0 + l];
        b.y = sB[(ka + 1) * 64 + n0 + l];
        acc = __builtin_amdgcn_wmma_f32_16x16x4_f32(false, a, false, b,
                                                    (short)0, acc, false, false);
    }

    const int rbase = m0 + mt * 16 + half * 8;
    if (m0 + 32 <= M) {                                // full tile: branch-free stores
        #pragma unroll
        for (int r = 0; r < 8; ++r)
            H[(rbase + r) * 64 + n0 + l] = acc[r];
    } else {                                           // remainder tile (never hit for N=100000)
        #pragma unroll
        for (int r = 0; r < 8; ++r) {
            int row = rbase + r;
            if (row < M) H[row * 64 + n0 + l] = acc[r];
        }
    }
}

// ---------------------------------------------------------------------------
// GEMM2: G[M,128] = A[M,64] @ W2[64,128]   (16 WMMA steps over K)
// 256 threads = 8 waves; block = one 16-row m-tile, wave w -> n-tile w.
// W2 (32 KB) is staged into LDS with GLOBAL_LOAD_ASYNC_TO_LDS_B128 (ASYNCcnt).
// ---------------------------------------------------------------------------
__global__ void gemm2_wmma_kernel(const float* __restrict__ A,
                                  const float* __restrict__ W2,
                                  float* __restrict__ G, int M) {
    __shared__ float sA[16 * 64];     // 4 KB
    __shared__ float sB[64 * 128];    // 32 KB (entire W2)
    const int m0 = blockIdx.x * 16;
    const int t = threadIdx.x;

    // --- async staging of W2 into LDS: 256 lanes x 8 iters x 16 B = 32 KB ---
    {
        // low 32 bits of a flat shared pointer == wave-relative LDS byte offset
        unsigned lds_base = (unsigned)(unsigned long long)(void*)&sB[0];
        #pragma unroll
        for (int it = 0; it < 8; ++it) {
            unsigned idx = (unsigned)(it * 256 + t) * 16u;   // byte offset
            unsigned lds_addr = lds_base + idx;
            asm volatile("global_load_async_to_lds_b128 %0, %1, %2"
                         :
                         : "v"(lds_addr), "v"(idx), "s"(W2)
                         : "memory");
        }
        asm volatile("s_wait_asynccnt 0" ::: "memory");
    }

    // --- A tile via normal LDS stores ---
    for (int i = t; i < 16 * 64; i += 256) {
        int r = i >> 6, c = i & 63;
        int row = m0 + r; if (row >= M) row = M - 1;
        sA[i] = A[row * 64 + c];
    }
    __syncthreads();

    const int wave = t >> 5, lane = t & 31;
    const int n0 = wave * 16;
    const int half = lane >> 4, l = lane & 15;

    v8f acc = {};
    #pragma unroll
    for (int kk = 0; kk < 64; kk += 4) {
        int ka = kk + half * 2;
        v2f a, b;
        a.x = sA[l * 64 + ka];
        a.y = sA[l * 64 + ka + 1];
        b.x = sB[ka * 128 + n0 + l];
        b.y = sB[(ka + 1) * 128 + n0 + l];
        acc = __builtin_amdgcn_wmma_f32_16x16x4_f32(false, a, false, b,
                                                    (short)0, acc, false, false);
    }

    const int rbase = m0 + half * 8;
    if (m0 + 16 <= M) {                                // full tile: branch-free stores
        #pragma unroll
        for (int r = 0; r < 8; ++r)
            G[(rbase + r) * 128 + n0 + l] = acc[r];
    } else {
        #pragma unroll
        for (int r = 0; r < 8; ++r) {
            int row = rbase + r;
            if (row < M) G[row * 128 + n0 + l] = acc[r];
        }
    }
}

// ---------------------------------------------------------------------------
// Aggregation: out[i] = sum_e norm(e) * h[row(e)]  over edges with col(e)=i,
// plus self loop (norm = dinv[i]^2) and bias, done in the init pass.
// ---------------------------------------------------------------------------
template <int F>
__global__ void agg_init_kernel(const float* __restrict__ h,
                                const float* __restrict__ dinv,
                                const float* __restrict__ bias,
                                float* __restrict__ out, int n) {
    long long i = (long long)blockIdx.x * blockDim.x + threadIdx.x;
    long long total = (long long)n * F;
    if (i < total) {
        int node = (int)(i / F);
        int f = (int)(i - (long long)node * F);
        float d = dinv[node];
        out[i] = h[i] * d * d + bias[f];
    }
}

// 16 threads per edge, each handles one float4 of the 64-wide feature row
__global__ void agg_edges64_kernel(const long long* __restrict__ ei,
                                   const float* __restrict__ w,
                                   const float* __restrict__ dinv,
                                   const float* __restrict__ h,
                                   float* __restrict__ out, int E) {
    long long t = (long long)blockIdx.x * blockDim.x + threadIdx.x;
    int e = (int)(t >> 4);
    int g = (int)(t & 15);
    if (e >= E) return;
    int r = (int)ei[e];
    int c = (int)ei[(long long)E + e];
    float norm = dinv[r] * w[e] * dinv[c];
    float4 v = ((const float4*)(h + (long long)r * 64))[g];
    float* oc = out + (long long)c * 64 + g * 4;
    atomicAdd(oc + 0, v.x * norm);
    atomicAdd(oc + 1, v.y * norm);
    atomicAdd(oc + 2, v.z * norm);
    atomicAdd(oc + 3, v.w * norm);
}

// 32 threads (one wave) per edge, each handles one float4 of 128-wide row
__global__ void agg_edges128_kernel(const long long* __restrict__ ei,
                                    const float* __restrict__ w,
                                    const float* __restrict__ dinv,
                                    const float* __restrict__ h,
                                    float* __restrict__ out, int E) {
    long long t = (long long)blockIdx.x * blockDim.x + threadIdx.x;
    int e = (int)(t >> 5);
    int g = (int)(t & 31);
    if (e >= E) return;
    int r = (int)ei[e];
    int c = (int)ei[(long long)E + e];
    float norm = dinv[r] * w[e] * dinv[c];
    float4 v = ((const float4*)(h + (long long)r * 128))[g];
    float* oc = out + (long long)c * 128 + g * 4;
    atomicAdd(oc + 0, v.x * norm);
    atomicAdd(oc + 1, v.y * norm);
    atomicAdd(oc + 2, v.z * norm);
    atomicAdd(oc + 3, v.w * norm);
}

// ---------------------------------------------------------------------------
// BatchNorm (training-mode stats, biased variance) folded to scale/shift
// one block per channel (64 channels)
// ---------------------------------------------------------------------------
__global__ void bn_stats_kernel(const float* __restrict__ a,
                                const float* __restrict__ gamma,
                                const float* __restrict__ beta,
                                float* __restrict__ scale,
                                float* __restrict__ shift, int n) {
    const int c = blockIdx.x;
    const int t = threadIdx.x;
    double s = 0.0, s2 = 0.0;
    for (int i = t; i < n; i += blockDim.x) {
        float v = a[(long long)i * 64 + c];
        s += v;
        s2 += (double)v * v;
    }
    __shared__ double sh[256], sh2[256];
    sh[t] = s; sh2[t] = s2;
    __syncthreads();
    for (int o = 128; o > 0; o >>= 1) {
        if (t < o) { sh[t] += sh[t + o]; sh2[t] += sh2[t + o]; }
        __syncthreads();
    }
    if (t == 0) {
        double mean = sh[0] / n;
        double var = sh2[0] / n - mean * mean;
        float inv = rsqrtf((float)var + 1e-5f);
        float sc = gamma[c] * inv;
        scale[c] = sc;
        shift[c] = beta[c] - (float)mean * sc;
    }
}

__global__ void bn_relu_kernel(float* __restrict__ a,
                               const float* __restrict__ scale,
                               const float* __restrict__ shift, int n) {
    long long i = (long long)blockIdx.x * blockDim.x + threadIdx.x;
    long long total = (long long)n * 64;
    if (i < total) {
        int f = (int)(i & 63);
        float v = a[i] * scale[f] + shift[f];
        a[i] = fmaxf(v, 0.0f);
    }
}

// ---------------------------------------------------------------------------
// Host-side launcher
// ---------------------------------------------------------------------------
extern "C" void kernel_launch(void* const* d_in, const int* in_sizes, int n_in,
                              void* d_out, int out_size, void* d_ws, size_t ws_size,
                              hipStream_t stream) {
    const float*      x   = (const float*)d_in[0];        // [N,11]
    const long long*  ei  = (const long long*)d_in[1];    // [2,E] int64
    const float*      ew  = (const float*)d_in[2];        // [E]
    const float*      W1  = (const float*)d_in[3];        // [11,64]
    const float*      b1  = (const float*)d_in[4];        // [64]
    const float*      gam = (const float*)d_in[5];        // [64]
    const float*      bet = (const float*)d_in[6];        // [64]
    const float*      W2  = (const float*)d_in[7];        // [64,128]
    const float*      b2  = (const float*)d_in[8];        // [128]
    float*            out = (float*)d_out;                // [N,128]

    const int N = in_sizes[0] / 11;
    const int E = in_sizes[2];

    // workspace layout (floats)
    float* ws    = (float*)d_ws;
    float* deg   = ws;                          // N
    float* dinv  = deg  + N;                    // N
    float* h1    = dinv + N;                    // N*64  (pre-aggregation)
    float* a1    = h1   + (size_t)N * 64;       // N*64  (post-agg / BN / ReLU)
    float* g2    = a1   + (size_t)N * 64;       // N*128 (pre-aggregation layer 2)
    float* scale = g2   + (size_t)N * 128;      // 64
    float* shift = scale + 64;                  // 64

    const int T = 256;
    auto cdiv = [](long long a, long long b) { return (int)((a + b - 1) / b); };

    // 1) degree + symmetric norm
    deg_init_kernel<<<cdiv(N, T), T, 0, stream>>>(deg, N);
    deg_scatter_kernel<<<cdiv(E, T), T, 0, stream>>>(ei, ew, deg, E);
    dinv_kernel<<<cdiv(N, T), T, 0, stream>>>(deg, dinv, N);

    // 2) h1 = x @ W1  (WMMA f32, 32 rows per block)
    gemm1_wmma_kernel<<<cdiv(N, 32), T, 0, stream>>>(x, W1, h1, N);

    // 3) aggregate layer 1: self-loop + bias init, then edge scatter
    agg_init_kernel<64><<<cdiv((long long)N * 64, T), T, 0, stream>>>(h1, dinv, b1, a1, N);
    agg_edges64_kernel<<<cdiv((long long)E * 16, T), T, 0, stream>>>(ei, ew, dinv, h1, a1, E);

    // 4) BatchNorm (training stats) + ReLU, in place on a1
    bn_stats_kernel<<<64, T, 0, stream>>>(a1, gam, bet, scale, shift, N);
    bn_relu_kernel<<<cdiv((long long)N * 64, T), T, 0, stream>>>(a1, scale, shift, N);

    // 5) g2 = a1 @ W2  (WMMA f32, async W2 staging, 16 rows per block)
    gemm2_wmma_kernel<<<cdiv(N, 16), T, 0, stream>>>(a1, W2, g2, N);

    // 6) aggregate layer 2 into d_out
    agg_init_kernel<128><<<cdiv((long long)N * 128, T), T, 0, stream>>>(g2, dinv, b2, out, N);
    agg_edges128_kernel<<<cdiv((long long)E * 32, T), T, 0, stream>>>(ei, ew, dinv, g2, out, E);
}